// LightGCN_81312320848128
// MI455X (gfx1250) — compile-verified
//
#include <hip/hip_runtime.h>
#include <stdint.h>

// ---------------------------------------------------------------------------
// LightGCN (3 layers) on MI455X / gfx1250.
// Memory-bound gather/scatter: no matmul structure (random indices), so the
// win comes from L2-resident ping-pong buffers, coalesced 256B row transfers,
// no-return global_atomic_add_f32 scatters, and async global->LDS staging of
// per-block interleaved edge records (GLOBAL_LOAD_ASYNC_TO_LDS_B32+ASYNCcnt).
// scatter_kernel is first in the file so the disasm snippet shows it.
// ---------------------------------------------------------------------------

#define DIMS 64
#define VEC 4
#define TPE 16              // threads per edge = DIMS/VEC
#define BLOCK 256
#define EPI 16              // edges per inner iteration = BLOCK/TPE
#define SCATTER_ITERS 4
#define EPB 64              // edges per block = EPI * SCATTER_ITERS
#define MWORDS 3            // record: {src, dst, w-bits}

#if defined(__HIP_DEVICE_COMPILE__) && defined(__gfx1250__)
#if __has_builtin(__builtin_amdgcn_global_load_async_to_lds_b32)
#define HAVE_ASYNC_LDS 1
#endif
#endif

__device__ __forceinline__ void atom_add_f32(float* p, float v) {
  // Hardware global_atomic_add_f32, no return value (STOREcnt, fire & forget).
  unsafeAtomicAdd(p, v);
}

#ifdef HAVE_ASYNC_LDS
typedef int __attribute__((address_space(1))) gint_t;
typedef int __attribute__((address_space(3))) lint_t;

__device__ __forceinline__ void async_g2l_b32(const void* g, void* l) {
  // generic -> AS1 / AS3 via uintptr round-trip. Low 32 bits of a generic LDS
  // pointer are the LDS byte offset per the CDNA5 aperture rules.
  gint_t* gp = reinterpret_cast<gint_t*>(reinterpret_cast<uintptr_t>(g));
  lint_t* lp = reinterpret_cast<lint_t*>(
      static_cast<uint32_t>(reinterpret_cast<uintptr_t>(l)));
  __builtin_amdgcn_global_load_async_to_lds_b32(gp, lp, /*offset=*/0,
                                                /*cpol=*/0);
}
#endif

__device__ __forceinline__ void wait_async_lds() {
#if defined(HAVE_ASYNC_LDS)
#if __has_builtin(__builtin_amdgcn_s_wait_asynccnt)
  __builtin_amdgcn_s_wait_asynccnt(0);
#else
  asm volatile("s_wait_asynccnt 0" ::: "memory");
#endif
#endif
}

// ---------------------------------------------------------------------------
// HOT KERNEL (placed first so the device-asm snippet shows it).
// One block handles EPB=64 edges. The block's contiguous 64x3-word metadata
// slab is staged into LDS with a single uniform async global->LDS b32 copy
// (thread t < 192 moves word t), then 16 lanes x float4 per edge gather a
// 256B row from x_cur and scatter via no-return global_atomic_add_f32.
// ---------------------------------------------------------------------------
__global__ __launch_bounds__(BLOCK) void scatter_kernel(
    const int* __restrict__ meta,   // MWORDS words per edge: src, dst, w-bits
    const float* __restrict__ xcur, float* __restrict__ xnext, int E) {
  __shared__ int s_meta[MWORDS * EPB];

  const int t = threadIdx.x;
  const long base = (long)blockIdx.x * EPB;
  const long mbase = base * MWORDS;
  const long mtotal = (long)E * MWORDS;

#ifdef HAVE_ASYNC_LDS
  {
    const long m = mbase + t;
    if (t < MWORDS * EPB && m < mtotal) async_g2l_b32(&meta[m], &s_meta[t]);
  }
  wait_async_lds();
  __syncthreads();
#else
  {
    const long m = mbase + t;
    if (t < MWORDS * EPB && m < mtotal) s_meta[t] = meta[m];
  }
  __syncthreads();
#endif

  const int slot = t >> 4;          // 0..15: edge slot within iteration
  const int col = (t & 15) * VEC;   // 0,4,...,60: column offset in the row

#pragma unroll
  for (int it = 0; it < SCATTER_ITERS; ++it) {
    const int el = slot + it * EPI;
    if (base + el < E) {
      const int s = s_meta[MWORDS * el + 0];
      const int d = s_meta[MWORDS * el + 1];
      const float ww = __int_as_float(s_meta[MWORDS * el + 2]);
      const float4 xv =
          *reinterpret_cast<const float4*>(xcur + (long)s * DIMS + col);
      float* op = xnext + (long)d * DIMS + col;
      atom_add_f32(op + 0, ww * xv.x);
      atom_add_f32(op + 1, ww * xv.y);
      atom_add_f32(op + 2, ww * xv.z);
      atom_add_f32(op + 3, ww * xv.w);
    }
  }
}

// ---------------------------------------------------------------------------
// Small utility kernels
// ---------------------------------------------------------------------------

__global__ void zero_f32_kernel(float* __restrict__ p, int n) {
  int i = blockIdx.x * blockDim.x + threadIdx.x;
  if (i < n) p[i] = 0.0f;
}

// deg[dst] += 1 for each edge (float atomic; counts are small integers).
__global__ void degree_kernel(const int* __restrict__ dst,
                              float* __restrict__ deg, int E) {
  int e = blockIdx.x * blockDim.x + threadIdx.x;
  if (e < E) atom_add_f32(&deg[dst[e]], 1.0f);
}

// In-place deg -> dinv = deg>0 ? rsqrt(deg) : 0
__global__ void dinv_kernel(float* __restrict__ deg_dinv, int N) {
  int i = blockIdx.x * blockDim.x + threadIdx.x;
  if (i < N) {
    float d = deg_dinv[i];
    deg_dinv[i] = (d > 0.0f) ? rsqrtf(d) : 0.0f;
  }
}

// meta[e] = {src[e], dst[e], bits(dinv[src]*dinv[dst])} interleaved records
__global__ void meta_kernel(const int* __restrict__ src,
                            const int* __restrict__ dst,
                            const float* __restrict__ dinv,
                            int* __restrict__ meta, int E) {
  int e = blockIdx.x * blockDim.x + threadIdx.x;
  if (e < E) {
    const int s = src[e];
    const int d = dst[e];
    const float w = dinv[s] * dinv[d];
    int* r = meta + (long)e * MWORDS;
    r[0] = s;
    r[1] = d;
    r[2] = __float_as_int(w);
  }
}

// out_lo = E0 (layer-0 x), acc = E0 (running layer sum), zero first ping-pong
__global__ void init_kernel(const float4* __restrict__ e0,
                            float4* __restrict__ out_lo,
                            float4* __restrict__ acc,
                            float4* __restrict__ zbuf, int n4) {
  int i = blockIdx.x * blockDim.x + threadIdx.x;
  if (i < n4) {
    float4 v = e0[i];
    out_lo[i] = v;
    acc[i] = v;
    zbuf[i] = make_float4(0.f, 0.f, 0.f, 0.f);
  }
}

// acc += x  (x = freshly scattered layer), and zero zbuf (the buffer the NEXT
// layer will scatter into; disjoint from x). Fuses two sweeps into one pass.
__global__ void acc_add_zero_kernel(float4* __restrict__ acc,
                                    const float4* __restrict__ x,
                                    float4* __restrict__ zbuf, int n4) {
  int i = blockIdx.x * blockDim.x + threadIdx.x;
  if (i < n4) {
    float4 a = acc[i];
    float4 b = x[i];
    a.x += b.x; a.y += b.y; a.z += b.z; a.w += b.w;
    acc[i] = a;
    if (zbuf) zbuf[i] = make_float4(0.f, 0.f, 0.f, 0.f);
  }
}

// out_hi = acc * 0.25 (in place), out_lo = E0 (restore exact copy)
__global__ void finalize_kernel(const float4* __restrict__ e0,
                                float4* __restrict__ out_lo,
                                float4* __restrict__ out_hi, int n4) {
  int i = blockIdx.x * blockDim.x + threadIdx.x;
  if (i < n4) {
    float4 a = out_hi[i];
    a.x *= 0.25f; a.y *= 0.25f; a.z *= 0.25f; a.w *= 0.25f;
    out_hi[i] = a;
    out_lo[i] = e0[i];
  }
}

// ---------------------------------------------------------------------------
// Host launch
// ---------------------------------------------------------------------------
extern "C" void kernel_launch(void* const* d_in, const int* in_sizes, int n_in,
                              void* d_out, int out_size, void* d_ws,
                              size_t ws_size, hipStream_t stream) {
  const float* E0 = (const float*)d_in[0];
  const int* ei = (const int*)d_in[1];  // harness passes integers as int32

  const int ND = in_sizes[0];      // N * 64
  const int N = ND / DIMS;
  const int E = in_sizes[1] / 2;   // edge_index is (2, E): src rows then dst
  const int* src = ei;
  const int* dst = ei + E;

  // Workspace layout (floats/ints): [dinv: N][xB: ND][meta: 3E]  (~38 MB)
  float* ws = (float*)d_ws;
  float* dinv = ws;
  float* xB = ws + N;
  int* meta = (int*)(xB + ND);

  float* out_lo = (float*)d_out;   // ends as E0 copy; used as ping-pong buf
  float* out_hi = out_lo + ND;     // running accumulator -> final mean

  const int n4 = ND / 4;
  const int gN = (N + BLOCK - 1) / BLOCK;
  const int gE = (E + BLOCK - 1) / BLOCK;
  const int g4 = (n4 + BLOCK - 1) / BLOCK;
  const int gScatter = (int)((E + EPB - 1) / EPB);

  // 1) degree -> dinv
  zero_f32_kernel<<<gN, BLOCK, 0, stream>>>(dinv, N);
  degree_kernel<<<gE, BLOCK, 0, stream>>>(dst, dinv, E);
  dinv_kernel<<<gN, BLOCK, 0, stream>>>(dinv, N);

  // 2) interleaved per-edge records {src, dst, norm-weight}
  meta_kernel<<<gE, BLOCK, 0, stream>>>(src, dst, dinv, meta, E);

  // 3) x0 = E0 (in out_lo), acc = E0 (in out_hi), zero first scatter target xB
  init_kernel<<<g4, BLOCK, 0, stream>>>((const float4*)E0, (float4*)out_lo,
                                        (float4*)out_hi, (float4*)xB, n4);

  // 4) three propagation layers, ping-pong between out_lo and xB.
  //    After each scatter: acc += result, and zero the buffer the NEXT layer
  //    scatters into (which is the buffer the current layer read from).
  float* cur = out_lo;
  float* nxt = xB;
  for (int layer = 0; layer < 3; ++layer) {
    scatter_kernel<<<gScatter, BLOCK, 0, stream>>>(meta, cur, nxt, E);
    float4* zb = (layer < 2) ? (float4*)cur : nullptr;  // cur becomes next tgt
    acc_add_zero_kernel<<<g4, BLOCK, 0, stream>>>((float4*)out_hi,
                                                  (const float4*)nxt, zb, n4);
    float* tmp = cur; cur = nxt; nxt = tmp;
  }

  // 5) mean over 4 layer embeddings; restore exact E0 copy in out_lo
  finalize_kernel<<<g4, BLOCK, 0, stream>>>((const float4*)E0,
                                            (float4*)out_lo, (float4*)out_hi,
                                            n4);
}